// DifferentiableMPCController_40888088658061
// MI455X (gfx1250) — compile-verified
//
#include <hip/hip_runtime.h>

typedef __attribute__((ext_vector_type(2))) float v2f;
typedef __attribute__((ext_vector_type(8))) float v8f;
typedef __attribute__((ext_vector_type(4))) unsigned int u32x4;
typedef __attribute__((ext_vector_type(8))) int i32x8;
typedef __attribute__((ext_vector_type(4))) int i32x4;

#define BB    512
#define THOR  100
#define NXC   24
#define NUC   8
#define NC    32
#define REGC  1e-6f
#define LD    33      // LDS tile row stride (32 + 1 pad, avoids bank conflicts)
#define AUGW  34      // augmented solve row stride

#if defined(__has_builtin)
#if __has_builtin(__builtin_amdgcn_tensor_load_to_lds) && __has_builtin(__builtin_amdgcn_s_wait_tensorcnt)
#define USE_TDM 1
#endif
#endif
#ifndef USE_TDM
#define USE_TDM 0
#endif

// ---- WMMA f32 16x16x4 wrapper (D = A(16x4) * B(4x16) + C) ----
__device__ __forceinline__ v8f wmma4(v2f a, v2f b, v8f c) {
  return __builtin_amdgcn_wmma_f32_16x16x4_f32(false, a, false, b, (short)0, c, false, false);
}

// A-fragment (16x4, MxK): lane l: m = l&15, K = (l>>4)*2 + r
__device__ __forceinline__ v2f afrag(const float* t, int mb, int kb, int h, int l16) {
  v2f a;
  a.x = t[(mb + l16) * LD + kb + 2 * h + 0];
  a.y = t[(mb + l16) * LD + kb + 2 * h + 1];
  return a;
}
// A-fragment reading the TRANSPOSE of an LDS tile: L[m][k] = t[k][m]
__device__ __forceinline__ v2f afragT(const float* t, int mb, int kb, int h, int l16) {
  v2f a;
  a.x = t[(kb + 2 * h + 0) * LD + mb + l16];
  a.y = t[(kb + 2 * h + 1) * LD + mb + l16];
  return a;
}
// B-fragment (4x16, KxN): lane l: n = l&15, K = (l>>4)*2 + r
__device__ __forceinline__ v2f bfrag(const float* t, int kb, int nb, int h, int l16) {
  v2f b;
  b.x = t[(kb + 2 * h + 0) * LD + nb + l16];
  b.y = t[(kb + 2 * h + 1) * LD + nb + l16];
  return b;
}
// C/D fragment store (16x16): M = (l>>4)*8 + r, N = l&15
__device__ __forceinline__ void store_cfrag(float* t, int mb, int nb, v8f c, int h, int l16) {
#pragma unroll
  for (int r = 0; r < 8; ++r) t[(mb + 8 * h + r) * LD + nb + l16] = c[r];
}

#if USE_TDM
// Async TDM copy of one 32x32 f32 stage-cost tile (HBM, row-major, stride 32)
// into an LDS tile with a 1-DWORD pad every 32 DWORDs (i.e. row stride LD=33).
// Tracked by TENSORcnt; TDM ops from one wave complete in order.
__device__ __forceinline__ void tdm_load_ct(const float* gsrc, float* ldst) {
  unsigned long long ga = (unsigned long long)(uintptr_t)gsrc;
  unsigned int la = (unsigned int)(uintptr_t)ldst;  // low 32 bits = LDS byte offset
  u32x4 g0;
  g0[0] = 1u;                                   // count=1 (valid user descriptor)
  g0[1] = la;                                   // lds_addr
  g0[2] = (unsigned int)ga;                     // global_addr[31:0]
  g0[3] = (unsigned int)(ga >> 32) | (2u << 30);// global_addr[56:32] | type=2
  i32x8 g1;
  g1[0] = (2 << 16)        // data_size = 4 bytes
        | (1 << 20)        // pad_enable
        | (4 << 22);       // pad_interval: 2^(4+1)=32 DWORDs  (pad_amount=0 -> 1 DWORD)
  g1[1] = (int)(32u << 16);  // tensor_dim0 = 32  (bits 63:48)
  g1[2] = (int)(32u << 16);  // tensor_dim1 = 32  (bits 95:80)
  g1[3] = (int)(32u << 16);  // tile_dim0  = 32  (bits 127:112)
  g1[4] = 32;                // tile_dim1  = 32, tile_dim2 = 0
  g1[5] = 32;                // tensor_dim0_stride = 32 (low 32 bits)
  g1[6] = 0;                 // stride0 hi, stride1 lo (unused for 2D)
  g1[7] = 0;
  i32x4 g2 = {1, 0, 0, 0};   // tensor_dim2 = 1 (benign for 2D tile)
  i32x4 g3 = {0, 0, 0, 0};
  i32x8 g4 = {0, 0, 0, 0, 0, 0, 0, 0};  // extra group (6-arg toolchain form)
  __builtin_amdgcn_tensor_load_to_lds(g0, g1, g2, g3, g4, 0);
}
#endif

// Fallback / non-TDM tile load: b128 vector loads + LDS stores.
__device__ __forceinline__ void vec_load_ct(const float* gsrc, float* ldst, int lane) {
  const float4* Ct4 = (const float4*)gsrc;
#pragma unroll
  for (int k2 = 0; k2 < 8; ++k2) {
    float4 f = Ct4[lane + k2 * 32];
    int pos = (lane + k2 * 32) << 2;
    float* p = &ldst[(pos >> 5) * LD + (pos & 31)];
    p[0] = f.x; p[1] = f.y; p[2] = f.z; p[3] = f.w;
  }
}

__global__ __launch_bounds__(32) void lqr_riccati_wmma(
    const float* __restrict__ x0,   const float* __restrict__ C,
    const float* __restrict__ cvec, const float* __restrict__ Cf,
    const float* __restrict__ cf,   const float* __restrict__ xref,
    const float* __restrict__ uref, const float* __restrict__ Adyn,
    const float* __restrict__ Bdyn, float* __restrict__ out) {
  // dynamic LDS: full gain schedule resident (100*(192+8) floats = 80 KB)
  extern __shared__ float sHist[];
  float* sKall  = sHist;                       // [THOR][NUC*NXC]
  float* skvall = sHist + THOR * NUC * NXC;    // [THOR][NUC]

  __shared__ float sA[NC * LD];     // A_dyn padded 32x32 (zeros beyond 24)
  __shared__ float sB[NC * LD];     // B_dyn padded (rows<24, cols<8 nonzero)
  __shared__ float sV[NC * LD];     // value Hessian V, padded
  __shared__ float sCtBuf[2][NC * LD];  // double-buffered C_t (TDM target; also Vn temp)
  __shared__ float sVA[NC * LD];    // V*A
  __shared__ float sVB[NC * LD];    // V*B (cols 0..15 used)
  __shared__ float sQux[16 * LD];   // Qux padded 16x32
  __shared__ float sK[16 * LD];     // K padded 16x32
  __shared__ float saug[NUC * AUGW];  // [Quu | Qux | Qu] 8x33
  __shared__ float srowk[AUGW];
  __shared__ float sfac[NUC];
  __shared__ float sq[NC], szr[NC], sQxv[NC], svv[NC], skv[NUC];
  __shared__ float sx[NC], su[NUC], sxn[NC];

  const int b = blockIdx.x;
  const int lane = threadIdx.x;
  const int h = lane >> 4;
  const int l16 = lane & 15;

  // ---- init padded A_dyn, B_dyn, V_T ----
  for (int idx = lane; idx < NC * NC; idx += 32) {
    int r = idx >> 5, c2 = idx & 31;
    sA[r * LD + c2] = (r < NXC && c2 < NXC) ? Adyn[r * NXC + c2] : 0.f;
    sB[r * LD + c2] = (r < NXC && c2 < NUC) ? Bdyn[r * NUC + c2] : 0.f;
    sV[r * LD + c2] = (r < NXC && c2 < NXC) ? Cf[(size_t)b * NC * NC + r * NC + c2] : 0.f;
  }
  szr[lane] = (lane < NXC) ? xref[((size_t)b * (THOR + 1) + THOR) * NXC + lane] : 0.f;
#if USE_TDM
  // kick off the first stage-cost tile asynchronously
  tdm_load_ct(C + ((size_t)b * THOR + (THOR - 1)) * NC * NC, sCtBuf[(THOR - 1) & 1]);
#endif
  __syncthreads();
  {  // v_T = c_final[:nx] - V_T @ x_ref[:, -1]
    float v = 0.f;
    if (lane < NXC) {
      v = cf[(size_t)b * NC + lane];
      for (int j = 0; j < NXC; ++j) v -= sV[lane * LD + j] * szr[j];
    }
    svv[lane] = v;
  }
  __syncthreads();

  // ================= backward Riccati scan =================
  for (int t = THOR - 1; t >= 0; --t) {
    float* sCt = sCtBuf[t & 1];
#if USE_TDM
    if (t > 0) {
      // issue next tile's DMA, then wait only for the older (current) one:
      // TDM ops from one wave complete in order -> tensorcnt<=1 means ours landed.
      tdm_load_ct(C + ((size_t)b * THOR + (t - 1)) * NC * NC, sCtBuf[(t - 1) & 1]);
      __builtin_amdgcn_s_wait_tensorcnt(1);
    } else {
      __builtin_amdgcn_s_wait_tensorcnt(0);
    }
#else
    vec_load_ct(C + ((size_t)b * THOR + t) * NC * NC, sCt, lane);
    if (t > 0) {  // prefetch next step's 4 KB slice (one 128B line per lane)
      const char* nxt = (const char*)(C + ((size_t)b * THOR + (t - 1)) * NC * NC);
      __builtin_prefetch(nxt + lane * 128, 0, 1);
    }
#endif
    szr[lane] = (lane < NXC) ? xref[((size_t)b * (THOR + 1) + t) * NXC + lane]
                             : uref[((size_t)b * THOR + t) * NUC + (lane - NXC)];
    __syncthreads();
    {  // q = c - C_t @ zref (one row per lane)
      float qv = cvec[((size_t)b * THOR + t) * NC + lane];
      for (int j = 0; j < NC; ++j) qv -= sCt[lane * LD + j] * szr[j];
      sq[lane] = qv;
    }
    __syncthreads();

    // ---- VA = V * A  (2x2 tiles, K=24) ----
#pragma unroll
    for (int mt = 0; mt < 2; ++mt)
#pragma unroll
      for (int nt = 0; nt < 2; ++nt) {
        v8f acc = {0.f, 0.f, 0.f, 0.f, 0.f, 0.f, 0.f, 0.f};
#pragma unroll
        for (int k8 = 0; k8 < 6; ++k8)
          acc = wmma4(afrag(sV, mt * 16, k8 * 4, h, l16),
                      bfrag(sA, k8 * 4, nt * 16, h, l16), acc);
        store_cfrag(sVA, mt * 16, nt * 16, acc, h, l16);
      }
    // ---- VB = V * B  (2x1 tiles) ----
#pragma unroll
    for (int mt = 0; mt < 2; ++mt) {
      v8f acc = {0.f, 0.f, 0.f, 0.f, 0.f, 0.f, 0.f, 0.f};
#pragma unroll
      for (int k8 = 0; k8 < 6; ++k8)
        acc = wmma4(afrag(sV, mt * 16, k8 * 4, h, l16),
                    bfrag(sB, k8 * 4, 0, h, l16), acc);
      store_cfrag(sVB, mt * 16, 0, acc, h, l16);
    }
    __syncthreads();

    // ---- Qxx = Cxx + A^T * VA (accumulators stay in VGPRs) ----
    v8f qxx[4];
#pragma unroll
    for (int mt = 0; mt < 2; ++mt)
#pragma unroll
      for (int nt = 0; nt < 2; ++nt) {
        v8f acc;
#pragma unroll
        for (int r = 0; r < 8; ++r) {
          int row = mt * 16 + 8 * h + r, col = nt * 16 + l16;
          acc[r] = (row < NXC && col < NXC) ? sCt[row * LD + col] : 0.f;
        }
#pragma unroll
        for (int k8 = 0; k8 < 6; ++k8)
          acc = wmma4(afragT(sA, mt * 16, k8 * 4, h, l16),
                      bfrag(sVA, k8 * 4, nt * 16, h, l16), acc);
        qxx[mt * 2 + nt] = acc;
      }
    // ---- Quu = Cuu + reg*I + B^T * VB  -> saug cols 0..7 ----
    {
      v8f acc;
#pragma unroll
      for (int r = 0; r < 8; ++r) {
        int m = 8 * h + r;
        acc[r] = (m < NUC && l16 < NUC)
                     ? (sCt[(NXC + m) * LD + NXC + l16] + (m == l16 ? REGC : 0.f))
                     : 0.f;
      }
#pragma unroll
      for (int k8 = 0; k8 < 6; ++k8)
        acc = wmma4(afragT(sB, 0, k8 * 4, h, l16),
                    bfrag(sVB, k8 * 4, 0, h, l16), acc);
      if (h == 0 && l16 < NUC) {
#pragma unroll
        for (int r = 0; r < 8; ++r) saug[r * AUGW + l16] = acc[r];
      }
    }
    // ---- Qux = Cxu^T + B^T * VA  (1x2 tiles) ----
#pragma unroll
    for (int nt = 0; nt < 2; ++nt) {
      v8f acc;
#pragma unroll
      for (int r = 0; r < 8; ++r) {
        int m = 8 * h + r, col = nt * 16 + l16;
        acc[r] = (m < NUC && col < NXC) ? sCt[(NXC + m) * LD + col] : 0.f;  // C symmetric
      }
#pragma unroll
      for (int k8 = 0; k8 < 6; ++k8)
        acc = wmma4(afragT(sB, 0, k8 * 4, h, l16),
                    bfrag(sVA, k8 * 4, nt * 16, h, l16), acc);
      store_cfrag(sQux, 0, nt * 16, acc, h, l16);
    }
    __syncthreads();

    // ---- Qx, Qu vectors (use previous v) ----
    if (lane < NXC) {
      float qx2 = sq[lane];
      for (int j = 0; j < NXC; ++j) qx2 += svv[j] * sA[j * LD + lane];
      sQxv[lane] = qx2;
    }
    if (lane < NUC) {
      float qu2 = sq[NXC + lane];
      for (int j = 0; j < NXC; ++j) qu2 += svv[j] * sB[j * LD + lane];
      saug[lane * AUGW + 32] = qu2;
    }
    for (int idx = lane; idx < NUC * NXC; idx += 32) {  // RHS cols 8..31
      int r = idx / NXC, c2 = idx - r * NXC;
      saug[r * AUGW + 8 + c2] = sQux[r * LD + c2];
    }
    __syncthreads();

    // ---- Gauss-Jordan on 8x33 augmented system (SPD + reg, no pivoting) ----
    for (int k = 0; k < NUC; ++k) {
      if (lane < NUC) sfac[lane] = saug[lane * AUGW + k];
      __syncthreads();
      float pinv = 1.f / sfac[k];
      if (lane < 33) srowk[lane] = saug[k * AUGW + lane] * pinv;
      __syncthreads();
      for (int idx = lane; idx < NUC * 33; idx += 32) {
        int r = idx / 33, c2 = idx - r * 33;
        saug[r * AUGW + c2] =
            (r == k) ? srowk[c2] : saug[r * AUGW + c2] - sfac[r] * srowk[c2];
      }
      __syncthreads();
    }

    // ---- K = -(Quu^-1 Qux), kv = -(Quu^-1 Qu) ----
    for (int idx = lane; idx < 16 * 32; idx += 32) {
      int r = idx >> 5, c2 = idx & 31;
      sK[r * LD + c2] = (r < NUC && c2 < NXC) ? -saug[r * AUGW + 8 + c2] : 0.f;
    }
    if (lane < NUC) skv[lane] = -saug[lane * AUGW + 32];
    __syncthreads();
    for (int idx = lane; idx < NUC * NXC; idx += 32) {  // save gain schedule
      int r = idx / NXC, c2 = idx - r * NXC;
      sKall[t * (NUC * NXC) + idx] = sK[r * LD + c2];
    }
    if (lane < NUC) skvall[t * NUC + lane] = skv[lane];

    // ---- Vn = Qxx + Qux^T * K  (K-dim = 8) -> reuse sCt as temp ----
#pragma unroll
    for (int mt = 0; mt < 2; ++mt)
#pragma unroll
      for (int nt = 0; nt < 2; ++nt) {
        v8f acc = qxx[mt * 2 + nt];
#pragma unroll
        for (int k8 = 0; k8 < 2; ++k8)
          acc = wmma4(afragT(sQux, mt * 16, k8 * 4, h, l16),
                      bfrag(sK, k8 * 4, nt * 16, h, l16), acc);
        store_cfrag(sCt, mt * 16, nt * 16, acc, h, l16);
      }
    __syncthreads();
    // symmetrize + re-pad into sV
    for (int idx = lane; idx < NC * NC; idx += 32) {
      int r = idx >> 5, c2 = idx & 31;
      sV[r * LD + c2] =
          (r < NXC && c2 < NXC) ? 0.5f * (sCt[r * LD + c2] + sCt[c2 * LD + r]) : 0.f;
    }
    // vn = Qx + Qux^T kv
    float vnew = 0.f;
    if (lane < NXC) {
      vnew = sQxv[lane];
      for (int j = 0; j < NUC; ++j) vnew += sQux[j * LD + lane] * skv[j];
    }
    __syncthreads();
    svv[lane] = (lane < NXC) ? vnew : 0.f;
    __syncthreads();
  }

  // ================= forward rollout (gains resident in LDS) =================
  if (lane < NXC) sx[lane] = x0[(size_t)b * NXC + lane];
  __syncthreads();
  float* outB = out + (size_t)b * ((THOR + 1) * NXC + THOR * NUC);
  for (int t = 0; t < THOR; ++t) {
    if (lane < NXC) outB[t * NXC + lane] = sx[lane];
    if (lane < NUC) {
      float u = skvall[t * NUC + lane];
      const float* Kt = &sKall[t * (NUC * NXC) + lane * NXC];
      for (int j = 0; j < NXC; ++j) u += Kt[j] * sx[j];
      su[lane] = u;
      outB[(THOR + 1) * NXC + t * NUC + lane] = u;
    }
    __syncthreads();
    if (lane < NXC) {
      float xn = 0.f;
      for (int j = 0; j < NXC; ++j) xn += sA[lane * LD + j] * sx[j];
      for (int m = 0; m < NUC; ++m) xn += sB[lane * LD + m] * su[m];
      sxn[lane] = xn;
    }
    __syncthreads();
    if (lane < NXC) sx[lane] = sxn[lane];
    __syncthreads();
  }
  if (lane < NXC) outB[THOR * NXC + lane] = sx[lane];
}

extern "C" void kernel_launch(void* const* d_in, const int* in_sizes, int n_in,
                              void* d_out, int out_size, void* d_ws, size_t ws_size,
                              hipStream_t stream) {
  const float* x0   = (const float*)d_in[0];
  const float* C    = (const float*)d_in[1];
  const float* cvec = (const float*)d_in[2];
  const float* Cf   = (const float*)d_in[3];
  const float* cf   = (const float*)d_in[4];
  const float* xref = (const float*)d_in[5];
  const float* uref = (const float*)d_in[6];
  const float* Adyn = (const float*)d_in[7];
  const float* Bdyn = (const float*)d_in[8];
  size_t shmem = (size_t)(THOR * NUC * NXC + THOR * NUC) * sizeof(float);  // 80 KB
  lqr_riccati_wmma<<<dim3(BB), dim3(32), shmem, stream>>>(
      x0, C, cvec, Cf, cf, xref, uref, Adyn, Bdyn, (float*)d_out);
  (void)in_sizes; (void)n_in; (void)out_size; (void)d_ws; (void)ws_size;
}